// SampledSoftmaxCrossEntropy_55568286876204
// MI455X (gfx1250) — compile-verified
//
#include <hip/hip_runtime.h>
#include <math.h>

typedef float v2f __attribute__((ext_vector_type(2)));
typedef float v8f __attribute__((ext_vector_type(8)));

#define NB     16384   // batch rows
#define DIM    768     // feature dim
#define NVOC   50257   // vocab
#define NSAMP  100     // sampled classes
#define NPAD   112     // padded to 7 tiles of 16
#define NTILES 7

// workspace layout (floats):
//   ws[0] = loss sum, ws[1] = valid count
//   ws[16 .. 16+112)          = gathered bias
//   ws[256 .. 256+112*768)    = gathered sampled projection rows (row-major)
#define WS_BIAS 16
#define WS_PROJ 256

__global__ __launch_bounds__(256)
void sce_gather_kernel(const float* __restrict__ projection,
                       const float* __restrict__ bias,
                       const int*   __restrict__ samples,
                       float* ws) {
    const int j = blockIdx.x;               // padded sample row 0..111
    if (j == 0 && threadIdx.x < 2) ws[threadIdx.x] = 0.0f;   // zero accumulators
    const int s = (j < NSAMP) ? samples[j] : -1;
    if (threadIdx.x == 0) ws[WS_BIAS + j] = (s >= 0) ? bias[s] : 0.0f;
    float* dst = ws + WS_PROJ + (size_t)j * DIM;
    const float* src = (s >= 0) ? (projection + (size_t)s * DIM) : nullptr;
    for (int c = threadIdx.x; c < DIM; c += 256)
        dst[c] = (s >= 0) ? src[c] : 0.0f;
}

__global__ __launch_bounds__(128)
void sce_main_kernel(const float* __restrict__ pred,
                     const int*   __restrict__ labels,
                     const float* __restrict__ projection,
                     const float* __restrict__ bias,
                     const int*   __restrict__ samples,
                     float* ws) {
    __shared__ float sScores[4][16][NPAD];   // per-wave 16x112 score tile
    __shared__ float sLabelScore[4][16];
    __shared__ int   sLabel[4][16];

    const int lane = threadIdx.x & 31;
    const int wave = threadIdx.x >> 5;
    const int half = lane >> 4;              // 0: K0/K1 pair, 1: K2/K3 pair
    const int m16  = lane & 15;
    const int row0 = blockIdx.x * 64 + wave * 16;

    const float* projG = ws + WS_PROJ;
    const float* biasG = ws + WS_BIAS;

    // Label row for the prediction row this lane carries through the K loop.
    const int myLabel  = labels[row0 + m16];
    const int safeLbl  = (myLabel >= 0) ? myLabel : 0;

    // ---------------- fp32 WMMA GEMM (16x112x768) + fused label dot ----------------
    v8f acc[NTILES] = {};
    float lacc = 0.0f;                       // partial pred[row]·proj[label] (this half's K)
    const float* aBase = pred       + (size_t)(row0 + m16) * DIM + 2 * half;
    const float* lBase = projection + (size_t)safeLbl      * DIM + 2 * half;
    #pragma unroll 2
    for (int k = 0; k < DIM; k += 4) {
        // A 16x4 (MxK): lane<16 holds K0,K1 of row M=lane; lane>=16 holds K2,K3
        v2f a = *(const v2f*)(aBase + k);
        v2f pl = *(const v2f*)(lBase + k);
        lacc = fmaf(a.x, pl.x, fmaf(a.y, pl.y, lacc));
        #pragma unroll
        for (int t = 0; t < NTILES; ++t) {
            // B 4x16 (KxN): N = lane&15, K pair selected by lane half
            v2f b = *(const v2f*)(projG + (size_t)(t * 16 + m16) * DIM + 2 * half + k);
            acc[t] = __builtin_amdgcn_wmma_f32_16x16x4_f32(
                false, a, false, b, (short)0, acc[t], false, false);
        }
    }
    // combine the two K-halves of the label dot (lanes m and m+16)
    lacc += __shfl_xor(lacc, 16, 32);
    if (lane < 16) {
        sLabel[wave][m16]      = myLabel;
        sLabelScore[wave][m16] = (myLabel >= 0) ? (lacc + bias[safeLbl]) : 0.0f;
    }

    // C/D layout: VGPR v, lanes 0-15 -> M=v, lanes 16-31 -> M=8+v, N=lane&15
    #pragma unroll
    for (int t = 0; t < NTILES; ++t) {
        #pragma unroll
        for (int v = 0; v < 8; ++v)
            sScores[wave][half * 8 + v][t * 16 + m16] = acc[t][v];
    }
    __syncthreads();

    // ---------------- per-row sampled-softmax CE ----------------
    if (threadIdx.x < 64) {
        const int w = threadIdx.x >> 4;
        const int i = threadIdx.x & 15;
        const int l = sLabel[w][i];
        if (l != -100) {
            const float logVm1 = __logf((float)(NVOC - 1));
            const float s0 = sLabelScore[w][i];
            int   rej  = 0;
            float mraw = -3.0e38f;
            for (int c = 0; c < NSAMP; ++c) {
                float raw = sScores[w][i][c] + biasG[c] + logVm1;
                if (samples[c] == l) { raw -= 1000000.0f; ++rej; }
                mraw = fmaxf(mraw, raw);
                sScores[w][i][c] = raw;           // stash adjusted score
            }
            const float tl = __logf((float)(NSAMP - rej));
            const float mx = fmaxf(mraw - tl, s0);
            float se = __expf(s0 - mx);
            for (int c = 0; c < NSAMP; ++c)
                se += __expf(sScores[w][i][c] - tl - mx);
            const float per_tok = mx + __logf(se) - s0;   // logsumexp - s0
            atomicAdd(&ws[0], per_tok);
            atomicAdd(&ws[1], 1.0f);
        }
    }
}

__global__ void sce_finalize_kernel(const float* ws, float* out) {
    if (threadIdx.x == 0)
        out[0] = ws[0] / fmaxf(ws[1], 1.0f);
}

extern "C" void kernel_launch(void* const* d_in, const int* in_sizes, int n_in,
                              void* d_out, int out_size, void* d_ws, size_t ws_size,
                              hipStream_t stream) {
    const float* pred    = (const float*)d_in[0];
    const int*   labels  = (const int*)  d_in[1];
    const float* proj    = (const float*)d_in[2];
    const float* bias    = (const float*)d_in[3];
    const int*   samples = (const int*)  d_in[4];
    float* ws  = (float*)d_ws;
    float* out = (float*)d_out;

    sce_gather_kernel  <<<NPAD,    256, 0, stream>>>(proj, bias, samples, ws);
    sce_main_kernel    <<<NB / 64, 128, 0, stream>>>(pred, labels, proj, bias, samples, ws);
    sce_finalize_kernel<<<1,        32, 0, stream>>>(ws, out);
}